// pLayer_1778116461268
// MI455X (gfx1250) — compile-verified
//
#include <hip/hip_runtime.h>
#include <hip/hip_bf16.h>
#include <math.h>

typedef float v2f __attribute__((ext_vector_type(2)));
typedef float v8f __attribute__((ext_vector_type(8)));

#define GMAXC 1.0f
#define GMINC 0.01f
#define EPSC  0.1f

static constexpr int N_ = 8, B_ = 32, NIN_ = 128, NOUT_ = 128, E_ = 1024, M_ = 130;
static constexpr int VSTR = 132;   // LDS row stride (dwords): even, bank-conflict-free, 8B aligned at even k

// ---------------------------------------------------------------------------
// Stage 1: per (n,o) build effective signed weights V[n][o][m] = tn/denom and
// bias c[n][o]. Tiny (1024 threads), entirely cache-resident.
// ---------------------------------------------------------------------------
__device__ __forceinline__ float theta_eff(float th) {
    th = fminf(fmaxf(th, -GMAXC), GMAXC);
    return (fabsf(th) < GMINC) ? 0.0f : th;
}

__global__ void prep_kernel(const float* __restrict__ theta,
                            const float* __restrict__ noise,
                            float* __restrict__ Vt,
                            float* __restrict__ cbias)
{
    int t = blockIdx.x * blockDim.x + threadIdx.x;   // 0..1023
    int n = t >> 7;
    int o = t & 127;
    const float* nz = noise + (size_t)n * M_ * NOUT_ + o;

    float denom = 1e-10f;
    for (int m = 0; m < M_; ++m) {
        float th = theta_eff(theta[m * NOUT_ + o]);
        float f  = nz[(size_t)m * NOUT_];
        float tn = th * ((f * 2.0f - 1.0f) * EPSC + 1.0f);
        denom += fabsf(tn);
    }
    float inv = 1.0f / denom;

    float bias = 0.0f;
    float* vrow = Vt + ((size_t)n * NOUT_ + o) * NIN_;
    for (int m = 0; m < M_; ++m) {
        float th = theta_eff(theta[m * NOUT_ + o]);
        float f  = nz[(size_t)m * NOUT_];
        float tn = th * ((f * 2.0f - 1.0f) * EPSC + 1.0f);
        if (m < NIN_) {
            vrow[m] = tn * inv;              // Wpos - Wneg  (signed, normalized)
            bias += fmaxf(-tn, 0.0f);        // Σ Wneg over input rows (from 1-a term)
        } else if (m == NIN_) {
            bias += fmaxf(tn, 0.0f);         // Wpos of the "ones" row
        }                                    // m==129 contributes only to denom
    }
    cbias[n * NOUT_ + o] = bias * inv;
}

// ---------------------------------------------------------------------------
// Stage 2: per (n,b): OUT[o,e] = tanh-act( V[n]^T(128x128) * A(128x1024) + c ).
// f32 WMMA 16x16x4, one block = (n, b, 128-wide e-chunk, 64-row o-half).
// ---------------------------------------------------------------------------
__global__ void __launch_bounds__(256)
gemm_act_kernel(const float* __restrict__ a,
                const float* __restrict__ Vt,
                const float* __restrict__ cbias,
                const float* __restrict__ eta,
                float* __restrict__ out)
{
    __shared__ float ldsV[64 * VSTR];
    __shared__ float ldsC[64];

    const int bid    = blockIdx.x;
    const int ohalf  = bid & 1;
    const int echunk = (bid >> 1) & 7;
    const int b      = (bid >> 4) & 31;
    const int n      = bid >> 9;
    const int tid    = threadIdx.x;
    const int o0     = ohalf * 64;

    // Stage V half (64 rows x 128 k) into LDS, coalesced global reads.
    const float* vsrc = Vt + ((size_t)n * NOUT_ + o0) * NIN_;
    #pragma unroll
    for (int r = 0; r < 32; ++r) {
        int idx  = r * 256 + tid;
        int orow = idx >> 7;
        int m    = idx & 127;
        ldsV[orow * VSTR + m] = vsrc[(size_t)orow * NIN_ + m];
    }
    if (tid < 64) ldsC[tid] = cbias[n * NOUT_ + o0 + tid];
    __syncthreads();

    const int wave  = tid >> 5;
    const int lane  = tid & 31;
    const int lrow  = lane & 15;    // row/col index within 16
    const int lhalf = lane >> 4;    // selects K pair {0,1} vs {2,3}
    const int e0    = echunk * 128 + wave * 16;

    const float* Abase = a + ((size_t)(n * B_ + b) * NIN_) * E_ + e0 + lrow;

    v8f acc[4];
    #pragma unroll
    for (int ot = 0; ot < 4; ++ot)
        #pragma unroll
        for (int i = 0; i < 8; ++i) acc[ot][i] = 0.0f;

    for (int k0 = 0; k0 < NIN_; k0 += 4) {
        const int kk = k0 + 2 * lhalf;
        // B fragment (4 x 16): lane holds column e0+lrow, rows kk and kk+1.
        const float* ap = Abase + (size_t)kk * E_;
        v2f Bf;
        Bf.x = ap[0];
        Bf.y = ap[E_];
        #pragma unroll
        for (int ot = 0; ot < 4; ++ot) {
            // A fragment (16 x 4): lane holds row o = ot*16+lrow, K = kk, kk+1.
            const float* vp = &ldsV[(ot * 16 + lrow) * VSTR + kk];
            v2f Af;
            Af.x = vp[0];
            Af.y = vp[1];
            acc[ot] = __builtin_amdgcn_wmma_f32_16x16x4_f32(
                false, Af, false, Bf, (short)0, acc[ot], false, false);
        }
    }

    const float eta0 = eta[0], eta1 = eta[1], eta2 = eta[2], eta3 = eta[3];
    float* obase = out + ((size_t)(n * B_ + b) * NOUT_) * E_;
    #pragma unroll
    for (int ot = 0; ot < 4; ++ot) {
        #pragma unroll
        for (int r = 0; r < 8; ++r) {
            const int oo = ot * 16 + r + 8 * lhalf;   // D layout: VGPR r = row r (+8 upper half)
            float z   = acc[ot][r] + ldsC[oo];
            float act = eta0 + eta1 * tanhf((z - eta2) * eta3);
            obase[(size_t)(o0 + oo) * E_ + e0 + lrow] = act;
        }
    }
}

// ---------------------------------------------------------------------------
extern "C" void kernel_launch(void* const* d_in, const int* in_sizes, int n_in,
                              void* d_out, int out_size, void* d_ws, size_t ws_size,
                              hipStream_t stream)
{
    const float* a     = (const float*)d_in[0];   // [N,B,NIN,E]
    const float* theta = (const float*)d_in[1];   // [130,128]
    const float* noise = (const float*)d_in[2];   // [N,130,128]
    const float* eta   = (const float*)d_in[3];   // [4]
    float* out = (float*)d_out;                   // [N,B,NOUT,E]

    float* Vt = (float*)d_ws;                                 // N*128*128 floats
    float* cb = Vt + (size_t)N_ * NOUT_ * NIN_;               // N*128 floats

    prep_kernel<<<4, 256, 0, stream>>>(theta, noise, Vt, cb);

    const int nblocks = N_ * B_ * (E_ / 128) * 2;             // 4096
    gemm_act_kernel<<<nblocks, 256, 0, stream>>>(a, Vt, cb, eta, out);
}